// GraphConvolution_34308198761262
// MI455X (gfx1250) — compile-verified
//
#include <hip/hip_runtime.h>
#include <hip/hip_bf16.h>
#include <stddef.h>

// GCN layer for MI455X (gfx1250, wave32):
//   support = X @ W            (fp32 WMMA 16x16x4)
//   out     = bias (broadcast) (+=) scatter-add over COO edges
//
// N_NODES = 80000, N_EDGES = 1280000, D_IN = D_OUT = 64

#define N_NODES 80000
#define N_EDGES 1280000
#define DIM 64

typedef float v2f __attribute__((ext_vector_type(2)));
typedef float v8f __attribute__((ext_vector_type(8)));

// ---------------------------------------------------------------------------
// Kernel 1: out[n, :] = bias[:]   (float4 vectorized broadcast)
// Grid covers N_NODES * DIM / 4 float4 elements.
// ---------------------------------------------------------------------------
__global__ void gcn_init_bias(const float* __restrict__ bias,
                              float* __restrict__ out) {
    int tid = blockIdx.x * blockDim.x + threadIdx.x;   // one float4 per thread
    const int total = N_NODES * (DIM / 4);             // 1,280,000
    if (tid >= total) return;
    int c = (tid & (DIM / 4 - 1)) << 2;                // channel group 0..60
    float4 b = *(const float4*)(bias + c);
    ((float4*)out)[tid] = b;
}

// ---------------------------------------------------------------------------
// Kernel 2: support = X @ W via V_WMMA_F32_16X16X4_F32, one wave per 16x16 tile.
//
// fp32 WMMA operand layouts (ISA 7.12.2):
//   A 16x4 : lanes 0-15 -> M=lane,   VGPR{0,1} = K{0,1}
//            lanes 16-31-> M=lane-16,VGPR{0,1} = K{2,3}
//   B 4x16 : lanes 0-15 -> N=lane,   VGPR{0,1} = K{0,1}
//            lanes 16-31-> N=lane-16,VGPR{0,1} = K{2,3}
//   C/D 16x16: VGPR r, lanes 0-15 -> (M=r,   N=lane)
//                      lanes 16-31-> (M=r+8, N=lane-16)
// ---------------------------------------------------------------------------
__global__ void gcn_gemm_xw(const float* __restrict__ X,   // [N_NODES, DIM]
                            const float* __restrict__ W,   // [DIM, DIM]
                            float* __restrict__ S) {       // [N_NODES, DIM]
    const int wave = (blockIdx.x * blockDim.x + threadIdx.x) >> 5;
    const int lane = threadIdx.x & 31;

    const int nTiles_n = DIM / 16;          // 4
    const int mTile = wave / nTiles_n;      // 0..4999
    const int nTile = wave % nTiles_n;      // 0..3
    if (mTile >= N_NODES / 16) return;

    const int half = lane >> 4;             // 0: K{0,1}, 1: K{2,3}
    const int l    = lane & 15;             // M (for A) / N (for B) within tile
    const int m0 = mTile * 16;
    const int n0 = nTile * 16;

    const float* __restrict__ xrow = X + (size_t)(m0 + l) * DIM + 2 * half;
    const float* __restrict__ wcol = W + (n0 + l);

    v8f acc = {};
#pragma unroll
    for (int k = 0; k < DIM; k += 4) {
        // A fragment: 2 consecutive K values from this lane's row of X
        v2f a = *(const v2f*)(xrow + k);
        // B fragment: W[k+2*half][n], W[k+2*half+1][n] (stride-DIM scalar loads;
        // W is 16 KB and stays resident in cache)
        const int kk = k + 2 * half;
        v2f b;
        b.x = wcol[(size_t)kk * DIM];
        b.y = wcol[(size_t)(kk + 1) * DIM];
        acc = __builtin_amdgcn_wmma_f32_16x16x4_f32(
            /*neg_a=*/false, a, /*neg_b=*/false, b,
            /*c_mod=*/(short)0, acc, /*reuse_a=*/false, /*reuse_b=*/false);
    }

    // Store D: lane half selects M-offset of 8; 16 lanes cover contiguous N
    float* __restrict__ srow = S + (size_t)(m0 + 8 * half) * DIM + n0 + l;
#pragma unroll
    for (int r = 0; r < 8; ++r) {
        srow[(size_t)r * DIM] = acc[r];
    }
}

// ---------------------------------------------------------------------------
// Kernel 3: out[dst] += edge_weight * support[src]
// 16 threads per edge, 4 channels per thread (float4 gather, 4 fp32 atomics).
// support and out both live in the 192 MB L2, so gathers + atomics resolve
// there rather than at HBM.
// ---------------------------------------------------------------------------
__global__ void gcn_spmm_scatter(const float* __restrict__ support,
                                 const int* __restrict__ src,
                                 const int* __restrict__ dst,
                                 const float* __restrict__ ew,
                                 float* __restrict__ out) {
    long long tid = (long long)blockIdx.x * blockDim.x + threadIdx.x;
    int e = (int)(tid >> 4);           // edge index
    if (e >= N_EDGES) return;
    int c = (int)(tid & 15) << 2;      // channel group

    int s = src[e];
    int d = dst[e];
    float w = ew[e];

    float4 v = *(const float4*)(support + (size_t)s * DIM + c);
    float* o = out + (size_t)d * DIM + c;
    atomicAdd(o + 0, v.x * w);
    atomicAdd(o + 1, v.y * w);
    atomicAdd(o + 2, v.z * w);
    atomicAdd(o + 3, v.w * w);
}

// ---------------------------------------------------------------------------
extern "C" void kernel_launch(void* const* d_in, const int* in_sizes, int n_in,
                              void* d_out, int out_size, void* d_ws, size_t ws_size,
                              hipStream_t stream) {
    const float* x     = (const float*)d_in[0];            // [N_NODES, DIM]
    const int*   eidx  = (const int*)d_in[1];              // [2, N_EDGES] flat
    const float* ew    = (const float*)d_in[2];            // [N_EDGES]
    const float* wgt   = (const float*)d_in[3];            // [DIM, DIM]
    const float* bias  = (const float*)d_in[4];            // [DIM]
    float*       out   = (float*)d_out;                    // [N_NODES, DIM]
    float*       supp  = (float*)d_ws;                     // scratch: 20.48 MB

    const int* src = eidx;                                 // row 0
    const int* dst = eidx + N_EDGES;                       // row 1

    // 1) out = bias broadcast
    {
        int total = N_NODES * (DIM / 4);
        int block = 256;
        int grid = (total + block - 1) / block;
        gcn_init_bias<<<grid, block, 0, stream>>>(bias, out);
    }

    // 2) support = X @ W  (one wave per 16x16 tile; 5000*4 = 20000 waves)
    {
        int waves = (N_NODES / 16) * (DIM / 16);
        int block = 256;                                   // 8 waves per block
        int grid = (waves * 32 + block - 1) / block;
        gcn_gemm_xw<<<grid, block, 0, stream>>>(x, wgt, supp);
    }

    // 3) scatter-add messages
    {
        long long threads = (long long)N_EDGES * 16;
        int block = 256;
        long long grid = (threads + block - 1) / block;
        gcn_spmm_scatter<<<(int)grid, block, 0, stream>>>(supp, src, dst, ew, out);
    }
}